// MPNN_46007689675022
// MI455X (gfx1250) — compile-verified
//
#include <hip/hip_runtime.h>

// ---------------------------------------------------------------------------
// MPNN forward for MI455X (gfx1250, wave32).
//   * fp32-native V_WMMA_F32_16X16X4_F32 for all dense layers.
//   * K=64 GEMMs: B matrix (64x64 = 128 VGPRs of v2f fragments) register-
//     resident per wave (__launch_bounds__(256,1) lifts the VGPR cap so it
//     does NOT spill); A strip (32 VGPRs) preloaded as a clause, then 64
//     WMMAs issue as a dense burst. Waves grid-stride over 16-row strips.
//   * edge einsum factored as T[k] = h @ W_edge[k] (17 slices incl. bias),
//     65 MB -> L2-resident; per-edge gather + global_atomic_add_f32 scatter,
//     with global_prefetch warming the scatter destinations.
// ---------------------------------------------------------------------------

#define DEVFN __device__ __forceinline__

typedef __attribute__((ext_vector_type(2))) float v2f;
typedef __attribute__((ext_vector_type(8))) float v8f;

constexpr int NN = 15000;   // nodes
constexpr int NE = 30000;   // edges
constexpr int NG = 512;     // graphs
constexpr int NSTRIPS = (NN + 15) / 16;   // 938 row strips

// Workspace layout (float offsets)
constexpr long OFF_H0   = 0;                       // [NN*64]
constexpr long OFF_H    = OFF_H0   + (long)NN*64;  // [NN*64]  (h == hidden)
constexpr long OFF_T    = OFF_H    + (long)NN*64;  // [17*NN*64]
constexpr long OFF_AGGR = OFF_T    + 17L*NN*64;    // [NN*64]
constexpr long OFF_M    = OFF_AGGR + (long)NN*64;  // [NN*64]
constexpr long OFF_GI   = OFF_M    + (long)NN*64;  // [NN*192]
constexpr long OFF_GH   = OFF_GI   + (long)NN*192; // [NN*192]
constexpr long OFF_SUMS = OFF_GH   + (long)NN*192; // [NG*128]
constexpr long OFF_CNT  = OFF_SUMS + (long)NG*128; // [NG]
constexpr long OFF_RD   = OFF_CNT  + NG;           // [NG*128]
constexpr long OFF_SP   = OFF_RD   + (long)NG*128; // [NG*512]
constexpr long OFF_HMID = OFF_SP   + (long)NG*512; // [NG*128]

DEVFN v8f wmma4(v2f a, v2f b, v8f c) {
  // V_WMMA_F32_16X16X4_F32 : D = A(16x4,f32) * B(4x16,f32) + C(16x16,f32)
  return __builtin_amdgcn_wmma_f32_16x16x4_f32(
      /*neg_a=*/false, a, /*neg_b=*/false, b,
      /*c_mod=*/(short)0, c, /*reuse_a=*/false, /*reuse_b=*/false);
}

// --- register-resident B fragments for a K=64 GEMM -------------------------
// Bf[ks][t] = 4x16 fragment for K-step ks and 16-col tile t.
// TB=false: B row-major [64 x ldb]; TB=true: B = W[N x 64] accessed transposed.
template<int COLT, bool TB>
DEVFN void load_bfrag64(const float* __restrict__ B, int ldb, v2f (&Bf)[16][COLT]) {
  const int lane = threadIdx.x & 31;
  const int half = lane >> 4;
  const int l15  = lane & 15;
#pragma unroll
  for (int ks = 0; ks < 16; ++ks) {
    const int kk = ks*4 + 2*half;
#pragma unroll
    for (int t = 0; t < COLT; ++t) {
      const int n = t*16 + l15;
      if (TB) {
        Bf[ks][t].x = B[(long)n*ldb + kk];
        Bf[ks][t].y = B[(long)n*ldb + kk + 1];
      } else {
        Bf[ks][t].x = B[(long)kk*ldb + n];
        Bf[ks][t].y = B[(long)(kk + 1)*ldb + n];
      }
    }
  }
}

// One 16-row strip: preload full A strip (16 x v2f), then dense WMMA burst.
// Rows clamped -> EXEC stays all-ones (WMMA requirement).
template<int COLT>
DEVFN void strip_mm64(const float* __restrict__ A, int lda, int r0, int rows,
                      const v2f (&Bf)[16][COLT], v8f (&acc)[COLT]) {
  const int lane = threadIdx.x & 31;
  const int half = lane >> 4;
  const int l15  = lane & 15;
  int ar = r0 + l15; if (ar > rows - 1) ar = rows - 1;
  const float* Arow = A + (long)ar * lda;
  v2f Af[16];
#pragma unroll
  for (int ks = 0; ks < 16; ++ks) {
    Af[ks].x = Arow[ks*4 + 2*half];
    Af[ks].y = Arow[ks*4 + 2*half + 1];
  }
#pragma unroll
  for (int ks = 0; ks < 16; ++ks)
#pragma unroll
    for (int t = 0; t < COLT; ++t) acc[t] = wmma4(Af[ks], Bf[ks][t], acc[t]);
}

// --- streaming variant for the small head GEMMs (K = 128 / 512) ------------
template<int COLT, bool TB>
DEVFN void wmma_row_strip(const float* __restrict__ A, int lda,
                          const float* __restrict__ B, int ldb,
                          int r0, int rows, int K, v8f acc[COLT]) {
  const int lane = threadIdx.x & 31;
  const int half = lane >> 4;
  const int l15  = lane & 15;
  int ar = r0 + l15; if (ar > rows - 1) ar = rows - 1;
  const float* Arow = A + (long)ar * lda;
  for (int kk = 0; kk < K; kk += 4) {
    v2f a;
    a.x = Arow[kk + 2*half];
    a.y = Arow[kk + 2*half + 1];
#pragma unroll
    for (int t = 0; t < COLT; ++t) {
      const int n = t*16 + l15;
      v2f b;
      if (TB) {
        b.x = B[(long)n*ldb + kk + 2*half];
        b.y = B[(long)n*ldb + kk + 2*half + 1];
      } else {
        b.x = B[(long)(kk + 2*half)*ldb + n];
        b.y = B[(long)(kk + 2*half + 1)*ldb + n];
      }
      acc[t] = wmma4(a, b, acc[t]);
    }
  }
}

DEVFN float sigmoidf_(float x) { return 1.0f / (1.0f + __expf(-x)); }

// ---------------------------------------------------------------------------
__global__ void k_zero(float* __restrict__ p, long n) {
  long t = (long)blockIdx.x * blockDim.x + threadIdx.x;
  if (t < n) p[t] = 0.0f;
}

// h0 = relu(x @ W_proj + b_proj); h = h0
__global__ void __launch_bounds__(256, 1)
k_proj(const float* __restrict__ x, const float* __restrict__ W,
       const float* __restrict__ bias, float* __restrict__ h0, float* __restrict__ h) {
  v2f Bf[16][4];
  load_bfrag64<4, false>(W, 64, Bf);
  const int lane = threadIdx.x & 31, half = lane >> 4, l15 = lane & 15;
  const int w0 = blockIdx.x * 8 + (threadIdx.x >> 5);
  const int wstride = gridDim.x * 8;
  for (int rt = w0; rt < NSTRIPS; rt += wstride) {
    const int r0 = rt * 16;
    v8f acc[4]; for (int t = 0; t < 4; ++t) acc[t] = 0.0f;
    strip_mm64<4>(x, 64, r0, NN, Bf, acc);
#pragma unroll
    for (int j = 0; j < 8; ++j) {
      const int r = r0 + j + 8*half;
      if (r < NN)
#pragma unroll
        for (int t = 0; t < 4; ++t) {
          const int c = t*16 + l15;
          float v = fmaxf(acc[t][j] + bias[c], 0.0f);
          h0[(long)r*64 + c] = v;
          h [(long)r*64 + c] = v;
        }
    }
  }
}

// T[k] = h @ W_edge[k]  (k<16), T[16] = h @ b_edge.reshape(64,64)
__global__ void __launch_bounds__(256, 1)
k_edgeT(const float* __restrict__ h, const float* __restrict__ W_edge,
        const float* __restrict__ b_edge, float* __restrict__ T) {
  const int k = blockIdx.y;
  const float* B = (k < 16) ? (W_edge + (long)k*4096) : b_edge;
  v2f Bf[16][4];
  load_bfrag64<4, false>(B, 64, Bf);
  float* Tk = T + (long)k * NN * 64;
  const int lane = threadIdx.x & 31, half = lane >> 4, l15 = lane & 15;
  const int w0 = blockIdx.x * 8 + (threadIdx.x >> 5);
  const int wstride = gridDim.x * 8;
  for (int rt = w0; rt < NSTRIPS; rt += wstride) {
    const int r0 = rt * 16;
    v8f acc[4]; for (int t = 0; t < 4; ++t) acc[t] = 0.0f;
    strip_mm64<4>(h, 64, r0, NN, Bf, acc);
#pragma unroll
    for (int j = 0; j < 8; ++j) {
      const int r = r0 + j + 8*half;
      if (r < NN)
#pragma unroll
        for (int t = 0; t < 4; ++t) Tk[(long)r*64 + t*16 + l15] = acc[t][j];
    }
  }
}

// msg[e] = T[16][src] + sum_k ea[e,k]*T[k][src]; scatter-add into aggr[dst]
__global__ void k_msg(const float* __restrict__ T, const int* __restrict__ ei,
                      const float* __restrict__ ea, float* __restrict__ aggr) {
  const long t = (long)blockIdx.x * blockDim.x + threadIdx.x;
  const int e = (int)(t >> 5);
  if (e >= NE) return;
  const int lane = threadIdx.x & 31;
  const int s = ei[e];        // src row (wave-uniform)
  const int d = ei[NE + e];   // dst row (wave-uniform)
  const int c0 = lane, c1 = lane + 32;
  // warm the scatter-destination lines while the gather is in flight
  __builtin_prefetch(&aggr[(long)d*64 + c0], 1, 3);
  const float* eav = ea + (long)e * 16;
  const float* Tb = T + 16L*NN*64 + (long)s*64;
  float m0 = Tb[c0], m1 = Tb[c1];
#pragma unroll
  for (int k = 0; k < 16; ++k) {
    const float w = eav[k];                       // broadcast load
    const float* Tk = T + (long)k*NN*64 + (long)s*64;
    m0 = fmaf(w, Tk[c0], m0);
    m1 = fmaf(w, Tk[c1], m1);
  }
  atomicAdd(&aggr[(long)d*64 + c0], m0);
  atomicAdd(&aggr[(long)d*64 + c1], m1);
}

// m = relu(aggr + h @ W_root + b_conv)
__global__ void __launch_bounds__(256, 1)
k_conv(const float* __restrict__ h, const float* __restrict__ W,
       const float* __restrict__ bias, const float* __restrict__ aggr,
       float* __restrict__ m) {
  v2f Bf[16][4];
  load_bfrag64<4, false>(W, 64, Bf);
  const int lane = threadIdx.x & 31, half = lane >> 4, l15 = lane & 15;
  const int w0 = blockIdx.x * 8 + (threadIdx.x >> 5);
  const int wstride = gridDim.x * 8;
  for (int rt = w0; rt < NSTRIPS; rt += wstride) {
    const int r0 = rt * 16;
    v8f acc[4]; for (int t = 0; t < 4; ++t) acc[t] = 0.0f;
    strip_mm64<4>(h, 64, r0, NN, Bf, acc);
#pragma unroll
    for (int j = 0; j < 8; ++j) {
      const int r = r0 + j + 8*half;
      if (r < NN)
#pragma unroll
        for (int t = 0; t < 4; ++t) {
          const int c = t*16 + l15;
          float v = acc[t][j] + aggr[(long)r*64 + c] + bias[c];
          m[(long)r*64 + c] = fmaxf(v, 0.0f);
        }
    }
  }
}

// gi = m @ W_ih^T + b_ih  (y==0);  gh = h @ W_hh^T + b_hh  (y==1)
// z = column group (3 groups of 64 cols) so B stays register-resident.
__global__ void __launch_bounds__(256, 1)
k_gates(const float* __restrict__ m, const float* __restrict__ h,
        const float* __restrict__ W_ih, const float* __restrict__ W_hh,
        const float* __restrict__ b_ih, const float* __restrict__ b_hh,
        float* __restrict__ gi, float* __restrict__ gh) {
  const int which = blockIdx.y;
  const int cg    = blockIdx.z;               // 0..2 -> output cols cg*64..+63
  const float* A    = which ? h    : m;
  const float* W    = (which ? W_hh : W_ih) + (long)cg*64*64;  // rows cg*64..
  const float* bias = (which ? b_hh : b_ih) + cg*64;
  float* out        = (which ? gh   : gi);
  v2f Bf[16][4];
  load_bfrag64<4, true>(W, 64, Bf);           // transposed access
  const int lane = threadIdx.x & 31, half = lane >> 4, l15 = lane & 15;
  const int w0 = blockIdx.x * 8 + (threadIdx.x >> 5);
  const int wstride = gridDim.x * 8;
  for (int rt = w0; rt < NSTRIPS; rt += wstride) {
    const int r0 = rt * 16;
    v8f acc[4]; for (int t = 0; t < 4; ++t) acc[t] = 0.0f;
    strip_mm64<4>(A, 64, r0, NN, Bf, acc);
#pragma unroll
    for (int j = 0; j < 8; ++j) {
      const int r = r0 + j + 8*half;
      if (r < NN)
#pragma unroll
        for (int t = 0; t < 4; ++t) {
          const int c = t*16 + l15;
          out[(long)r*192 + cg*64 + c] = acc[t][j] + bias[c];
        }
    }
  }
}

// GRU cell update, in place on h (== hidden)
__global__ void k_gru(const float* __restrict__ gi, const float* __restrict__ gh,
                      float* __restrict__ h) {
  const long t = (long)blockIdx.x * blockDim.x + threadIdx.x;
  if (t >= (long)NN*64) return;
  const int n = (int)(t >> 6), c = (int)(t & 63);
  const long b = (long)n * 192;
  const float r  = sigmoidf_(gi[b + c]      + gh[b + c]);
  const float z  = sigmoidf_(gi[b + 64 + c] + gh[b + 64 + c]);
  const float nn = tanhf(gi[b + 128 + c] + r * gh[b + 128 + c]);
  const float hp = h[t];
  h[t] = (1.0f - z) * nn + z * hp;
}

// p_borylation[n] = h[n] . W_cls + b_cls
__global__ void k_cls(const float* __restrict__ h, const float* __restrict__ W,
                      const float* __restrict__ b, float* __restrict__ out) {
  const int n = blockIdx.x * blockDim.x + threadIdx.x;
  if (n >= NN) return;
  float acc = 0.0f;
  const float* hr = h + (long)n * 64;
#pragma unroll
  for (int c = 0; c < 64; ++c) acc = fmaf(hr[c], W[c], acc);
  out[n] = acc + b[0];
}

// segment-sum of cat([h0, h]) over batch + counts
__global__ void k_reduce(const float* __restrict__ h0, const float* __restrict__ h,
                         const int* __restrict__ batch, float* __restrict__ sums,
                         float* __restrict__ cnt) {
  const long t = (long)blockIdx.x * blockDim.x + threadIdx.x;
  if (t >= (long)NN*128) return;
  const int n = (int)(t >> 7), c = (int)(t & 127);
  const int g = batch[n];
  const float v = (c < 64) ? h0[(long)n*64 + c] : h[(long)n*64 + (c - 64)];
  atomicAdd(&sums[(long)g*128 + c], v);
  if (c == 0) atomicAdd(&cnt[g], 1.0f);
}

__global__ void k_readout(const float* __restrict__ sums, const float* __restrict__ cnt,
                          float* __restrict__ rd) {
  const int t = blockIdx.x * blockDim.x + threadIdx.x;
  if (t >= NG*128) return;
  const int g = t >> 7;
  rd[t] = sums[t] / fmaxf(cnt[g], 1.0f);
}

// sp = prelu(readout[512x128] @ W_sp[128x512] + b_sp)
__global__ void __launch_bounds__(256, 1)
k_sp(const float* __restrict__ rd, const float* __restrict__ W,
     const float* __restrict__ bias, const float* __restrict__ prelu_a,
     float* __restrict__ sp) {
  const int w = blockIdx.x * 8 + (threadIdx.x >> 5);   // 1024 waves
  const int rt = w >> 5, ct = w & 31;
  const int r0 = rt * 16, c0 = ct * 16;
  if (r0 >= NG) return;
  v8f acc[1]; acc[0] = 0.0f;
  wmma_row_strip<1, false>(rd, 128, W + c0, 512, r0, NG, 128, acc);
  const float a = prelu_a[0];
  const int lane = threadIdx.x & 31, half = lane >> 4, l15 = lane & 15;
#pragma unroll
  for (int j = 0; j < 8; ++j) {
    const int r = r0 + j + 8*half;
    const int c = c0 + l15;
    float v = acc[0][j] + bias[c];
    sp[(long)r*512 + c] = (v >= 0.0f) ? v : a * v;
  }
}

// hmid = relu(sp[512x512] @ W_y1[512x128] + b_y1)
__global__ void __launch_bounds__(256, 1)
k_y1(const float* __restrict__ sp, const float* __restrict__ W,
     const float* __restrict__ bias, float* __restrict__ hmid) {
  const int w = blockIdx.x * 8 + (threadIdx.x >> 5);   // 256 waves
  const int rt = w >> 3, ct = w & 7;
  const int r0 = rt * 16, c0 = ct * 16;
  if (r0 >= NG) return;
  v8f acc[1]; acc[0] = 0.0f;
  wmma_row_strip<1, false>(sp, 512, W + c0, 128, r0, NG, 512, acc);
  const int lane = threadIdx.x & 31, half = lane >> 4, l15 = lane & 15;
#pragma unroll
  for (int j = 0; j < 8; ++j) {
    const int r = r0 + j + 8*half;
    const int c = c0 + l15;
    hmid[(long)r*128 + c] = fmaxf(acc[0][j] + bias[c], 0.0f);
  }
}

// yield[g] = hmid[g] . W_y2 + b_y2
__global__ void k_y2(const float* __restrict__ hmid, const float* __restrict__ W,
                     const float* __restrict__ b, float* __restrict__ out) {
  const int g = blockIdx.x * blockDim.x + threadIdx.x;
  if (g >= NG) return;
  float acc = 0.0f;
  const float* hr = hmid + (long)g * 128;
#pragma unroll
  for (int c = 0; c < 128; ++c) acc = fmaf(hr[c], W[c], acc);
  out[g] = acc + b[0];
}

// ---------------------------------------------------------------------------
extern "C" void kernel_launch(void* const* d_in, const int* in_sizes, int n_in,
                              void* d_out, int out_size, void* d_ws, size_t ws_size,
                              hipStream_t stream) {
  const float* x      = (const float*)d_in[0];
  const int*   ei     = (const int*)  d_in[1];
  const float* ea     = (const float*)d_in[2];
  const int*   batch  = (const int*)  d_in[3];
  const float* W_proj = (const float*)d_in[4];
  const float* b_proj = (const float*)d_in[5];
  const float* W_edge = (const float*)d_in[6];
  const float* b_edge = (const float*)d_in[7];
  const float* W_root = (const float*)d_in[8];
  const float* b_conv = (const float*)d_in[9];
  const float* W_ih   = (const float*)d_in[10];
  const float* W_hh   = (const float*)d_in[11];
  const float* b_ih   = (const float*)d_in[12];
  const float* b_hh   = (const float*)d_in[13];
  const float* W_cls  = (const float*)d_in[14];
  const float* b_cls  = (const float*)d_in[15];
  const float* W_sp   = (const float*)d_in[16];
  const float* b_sp   = (const float*)d_in[17];
  const float* prelu  = (const float*)d_in[18];
  const float* W_y1   = (const float*)d_in[19];
  const float* b_y1   = (const float*)d_in[20];
  const float* W_y2   = (const float*)d_in[21];
  const float* b_y2   = (const float*)d_in[22];

  float* out  = (float*)d_out;               // [NN] p_borylation, then [NG] yield
  float* ws   = (float*)d_ws;
  float* h0   = ws + OFF_H0;
  float* h    = ws + OFF_H;
  float* T    = ws + OFF_T;
  float* aggr = ws + OFF_AGGR;
  float* m    = ws + OFF_M;
  float* gi   = ws + OFF_GI;
  float* gh   = ws + OFF_GH;
  float* sums = ws + OFF_SUMS;
  float* cnt  = ws + OFF_CNT;
  float* rd   = ws + OFF_RD;
  float* sp   = ws + OFF_SP;
  float* hmid = ws + OFF_HMID;

  // Grid-stride GEMMs: ~2 strips per wave (amortizes register-resident B).
  const int GB1 = 59;   // single-slice GEMMs: 472 waves over 938 strips
  const int GBE = 30;   // edgeT: 240 waves/slice x 17 slices = 4080 waves

  k_proj<<<GB1, 256, 0, stream>>>(x, W_proj, b_proj, h0, h);

  for (int step = 0; step < 3; ++step) {
    k_zero<<<3750, 256, 0, stream>>>(aggr, (long)NN*64);
    k_edgeT<<<dim3(GBE, 17), 256, 0, stream>>>(h, W_edge, b_edge, T);
    k_msg<<<3750, 256, 0, stream>>>(T, ei, ea, aggr);
    k_conv<<<GB1, 256, 0, stream>>>(h, W_root, b_conv, aggr, m);
    k_gates<<<dim3(GB1, 2, 3), 256, 0, stream>>>(m, h, W_ih, W_hh, b_ih, b_hh, gi, gh);
    k_gru<<<3750, 256, 0, stream>>>(gi, gh, h);
  }

  k_cls<<<(NN + 255) / 256, 256, 0, stream>>>(h, W_cls, b_cls, out);

  k_zero<<<(NG*128 + NG + 255) / 256, 256, 0, stream>>>(sums, (long)NG*128 + NG);
  k_reduce<<<((long)NN*128 + 255) / 256, 256, 0, stream>>>(h0, h, batch, sums, cnt);
  k_readout<<<(NG*128 + 255) / 256, 256, 0, stream>>>(sums, cnt, rd);
  k_sp<<<128, 256, 0, stream>>>(rd, W_sp, b_sp, prelu, sp);
  k_y1<<<32, 256, 0, stream>>>(sp, W_y1, b_y1, hmid);
  k_y2<<<2, 256, 0, stream>>>(hmid, W_y2, b_y2, out + NN);
}